// CommunityGAT_52063593562731
// MI455X (gfx1250) — compile-verified
//
#include <hip/hip_runtime.h>
#include <hip/hip_bf16.h>
#include <math.h>

// ---------------------------------------------------------------------------
// CommunityGAT forward for MI455X (gfx1250, wave32).
// GEMMs via V_WMMA_F32_16X16X4_F32: NT column tiles per wave (A reuse) with a
// register double-buffered (software-pipelined) K loop so loads for step k+4
// are in flight while WMMAs for step k execute.
// Edge softmax via 3 passes (atomic-max, atomic-exp-sum, atomic scatter);
// edge kernels templated on head count / channels -> shifts instead of idiv,
// message traffic via b128 loads.
// ---------------------------------------------------------------------------

typedef __attribute__((ext_vector_type(2))) float v2f;
typedef __attribute__((ext_vector_type(8))) float v8f;

#define GAT_INCH   128
#define GAT_HID    32
#define GAT_HEADS  8
#define GAT_EMB    64
#define NEG_SLOPE  0.2f

// ---------------------------------------------------------------------------
// GEMM: H[N,M] = X[N,K] @ W[K,M], f32, WMMA 16x16x4.
// One wave32 computes a 16 x (16*NT) strip. Nrows % 16 == 0 (50000=3125*16),
// K % 4 == 0, M % (16*NT) == 0. All branches wave-uniform -> EXEC all-ones
// at every WMMA (ISA 7.12 restriction).
// ---------------------------------------------------------------------------
template <int NT>
__global__ void gat_gemm_wmma_f32(const float* __restrict__ X,
                                  const float* __restrict__ W,
                                  float* __restrict__ H,
                                  int Nrows, int K, int M) {
    const int lane     = threadIdx.x & 31;
    const int wave_id  = (blockIdx.x * blockDim.x + threadIdx.x) >> 5;
    const int groups_m = M / (16 * NT);
    const int tile_r   = wave_id / groups_m;
    const int tile_c0  = (wave_id % groups_m) * NT;
    if (tile_r * 16 >= Nrows) return;          // wave-uniform guard

    // A 16x4 layout: lanes 0-15 -> K = {0,1}, lanes 16-31 -> K = {2,3}
    const int m_lane = lane & 15;
    const int khalf  = (lane >> 4) << 1;       // 0 or 2
    const int row    = tile_r * 16 + m_lane;

    const float* xp = X + (size_t)row * K + khalf;            // A: xp[k], xp[k+1]
    const float* wp = W + (size_t)khalf * M + tile_c0 * 16 + m_lane; // B base

    v8f acc[NT];
#pragma unroll
    for (int t = 0; t < NT; ++t) acc[t] = (v8f){};

    // ---- prologue: fragments for k = 0 ----
    v2f a;  a.x = xp[0];  a.y = xp[1];
    v2f b[NT];
#pragma unroll
    for (int t = 0; t < NT; ++t) { b[t].x = wp[t * 16]; b[t].y = wp[t * 16 + M]; }

    // ---- pipelined main loop: prefetch k, compute k-4 ----
    for (int k = 4; k < K; k += 4) {
        v2f an;  an.x = xp[k];  an.y = xp[k + 1];
        v2f bn[NT];
        const float* wpk = wp + (size_t)k * M;
#pragma unroll
        for (int t = 0; t < NT; ++t) { bn[t].x = wpk[t * 16]; bn[t].y = wpk[t * 16 + M]; }
#pragma unroll
        for (int t = 0; t < NT; ++t)
            acc[t] = __builtin_amdgcn_wmma_f32_16x16x4_f32(
                false, a, false, b[t], (short)0, acc[t], false, false);
        a = an;
#pragma unroll
        for (int t = 0; t < NT; ++t) b[t] = bn[t];
    }
    // ---- epilogue ----
#pragma unroll
    for (int t = 0; t < NT; ++t)
        acc[t] = __builtin_amdgcn_wmma_f32_16x16x4_f32(
            false, a, false, b[t], (short)0, acc[t], false, false);

    // C/D 16x16 f32 layout: lanes 0-15 VGPR r -> M=r; lanes 16-31 -> M=r+8
    const int rbase = tile_r * 16 + ((lane >> 4) << 3);
#pragma unroll
    for (int t = 0; t < NT; ++t) {
        const int col = (tile_c0 + t) * 16 + m_lane;
#pragma unroll
        for (int r = 0; r < 8; ++r)
            H[(size_t)(rbase + r) * M + col] = acc[t][r];
    }
}

// ---------------------------------------------------------------------------
// Per-(node,head): alpha_src/alpha_dst dot products + init softmax stats.
// Hm is [N, HH*C] row-major; (n*HH + h)*C indexes head h of node n.
// ---------------------------------------------------------------------------
template <int HH, int C>
__global__ void gat_alpha_init(const float* __restrict__ Hm,
                               const float* __restrict__ a_src,
                               const float* __restrict__ a_dst,
                               float* __restrict__ as, float* __restrict__ ad,
                               float* __restrict__ mmax, float* __restrict__ denom,
                               int N) {
    int idx = blockIdx.x * blockDim.x + threadIdx.x;   // n*HH + h
    if (idx >= N * HH) return;
    const int h = idx % HH;                            // HH: const -> mask
    const float4* hp = (const float4*)(Hm + (size_t)idx * C);
    const float4* ps = (const float4*)(a_src + h * C);
    const float4* pd = (const float4*)(a_dst + h * C);
    float ss = 0.f, sd = 0.f;
#pragma unroll
    for (int c = 0; c < (C >> 2); ++c) {
        float4 v = hp[c], s4 = ps[c], d4 = pd[c];
        ss = fmaf(v.x, s4.x, fmaf(v.y, s4.y, fmaf(v.z, s4.z, fmaf(v.w, s4.w, ss))));
        sd = fmaf(v.x, d4.x, fmaf(v.y, d4.y, fmaf(v.z, d4.z, fmaf(v.w, d4.w, sd))));
    }
    as[idx]    = ss;
    ad[idx]    = sd;
    mmax[idx]  = -INFINITY;
    denom[idx] = 0.f;
}

// Order-independent float atomic max via int/uint punning (valid for any mix
// of signs given -inf init; maps to native integer global atomics).
__device__ __forceinline__ void atomicMaxF(float* addr, float val) {
    if (val >= 0.f) atomicMax((int*)addr, __float_as_int(val));
    else            atomicMin((unsigned int*)addr, __float_as_uint(val));
}

__device__ __forceinline__ void edge_decode(long long e, int E,
                                            const int* __restrict__ src,
                                            const int* __restrict__ dst,
                                            int& s, int& d) {
    if (e < E) { s = src[e]; d = dst[e]; }
    else       { s = d = (int)(e - E); }      // implicit self-loops
}

template <int HH>
__device__ __forceinline__ float edge_score(const float* as, const float* ad,
                                            int s, int d, int h) {
    float v = as[s * HH + h] + ad[d * HH + h];
    return v > 0.f ? v : v * NEG_SLOPE;       // leaky_relu(0.2)
}

// Pass 1: segment max over destination
template <int HH>
__global__ void gat_edge_max(const int* __restrict__ src, const int* __restrict__ dst,
                             const float* __restrict__ as, const float* __restrict__ ad,
                             float* __restrict__ mmax, int E, int N) {
    long long idx = (long long)blockIdx.x * blockDim.x + threadIdx.x;
    long long tot = (long long)(E + N) * HH;
    if (idx >= tot) return;
    const int h = (int)(idx % HH);
    long long e = idx / HH;
    int s, d; edge_decode(e, E, src, dst, s, d);
    atomicMaxF(&mmax[d * HH + h], edge_score<HH>(as, ad, s, d, h));
}

// Pass 2: segment sum of exp(e - max)
template <int HH>
__global__ void gat_edge_sum(const int* __restrict__ src, const int* __restrict__ dst,
                             const float* __restrict__ as, const float* __restrict__ ad,
                             const float* __restrict__ mmax, float* __restrict__ denom,
                             int E, int N) {
    long long idx = (long long)blockIdx.x * blockDim.x + threadIdx.x;
    long long tot = (long long)(E + N) * HH;
    if (idx >= tot) return;
    const int h = (int)(idx % HH);
    long long e = idx / HH;
    int s, d; edge_decode(e, E, src, dst, s, d);
    float v = edge_score<HH>(as, ad, s, d, h);
    atomicAdd(&denom[d * HH + h], expf(v - mmax[d * HH + h]));
}

// Pass 3: weighted scatter of messages h[src]*alpha into agg[dst].
// b128 message loads; per-float global_atomic_add_f32 scatter.
template <int HH, int C>
__global__ void gat_edge_agg(const int* __restrict__ src, const int* __restrict__ dst,
                             const float* __restrict__ as, const float* __restrict__ ad,
                             const float* __restrict__ mmax, const float* __restrict__ denom,
                             const float* __restrict__ Hm, float* __restrict__ agg,
                             int E, int N) {
    long long idx = (long long)blockIdx.x * blockDim.x + threadIdx.x;
    long long tot = (long long)(E + N) * HH;
    if (idx >= tot) return;
    const int h = (int)(idx % HH);
    long long e = idx / HH;
    int s, d; edge_decode(e, E, src, dst, s, d);
    float v = edge_score<HH>(as, ad, s, d, h);
    float w = expf(v - mmax[d * HH + h]) / (denom[d * HH + h] + 1e-16f);
    const float4* hp = (const float4*)(Hm + ((size_t)s * HH + h) * C);
    float*        op = agg + ((size_t)d * HH + h) * C;
#pragma unroll
    for (int c4 = 0; c4 < (C >> 2); ++c4) {
        float4 m = hp[c4];
        atomicAdd(&op[c4 * 4 + 0], w * m.x);
        atomicAdd(&op[c4 * 4 + 1], w * m.y);
        atomicAdd(&op[c4 * 4 + 2], w * m.z);
        atomicAdd(&op[c4 * 4 + 3], w * m.w);
    }
}

// bias add + optional ELU (jax.nn.elu: x>0 ? x : expm1(x)); float4 lanes.
template <int M4, int DO_ELU>
__global__ void gat_bias_act(const float* __restrict__ agg,
                             const float* __restrict__ bias,
                             float* __restrict__ out, long long total4) {
    long long i = (long long)blockIdx.x * blockDim.x + threadIdx.x;
    if (i >= total4) return;
    float4 v = ((const float4*)agg)[i];
    float4 b = ((const float4*)bias)[(int)(i % M4)];
    v.x += b.x; v.y += b.y; v.z += b.z; v.w += b.w;
    if (DO_ELU) {
        v.x = v.x > 0.f ? v.x : expm1f(v.x);
        v.y = v.y > 0.f ? v.y : expm1f(v.y);
        v.z = v.z > 0.f ? v.z : expm1f(v.z);
        v.w = v.w > 0.f ? v.w : expm1f(v.w);
    }
    ((float4*)out)[i] = v;
}

// ---------------------------------------------------------------------------
// Host-side orchestration (stream-only; graph-capture safe)
// ---------------------------------------------------------------------------
static inline int cdiv_i(long long a, int b) { return (int)((a + b - 1) / b); }

template <int HH, int C, int NT, int DO_ELU>
static void run_layer(hipStream_t stream,
                      const float* xin, const float* W, const float* a_s,
                      const float* a_d, const float* bias,
                      float* hbuf, float* aggbuf, float* stats,
                      float* out, int N, int E,
                      const int* src, const int* dst, int K) {
    constexpr int M = HH * C;
    // stats region: as | ad | mmax | denom, each N*HH floats
    float* as    = stats;
    float* ad    = stats + (size_t)N * HH;
    float* mmax  = stats + (size_t)2 * N * HH;
    float* denom = stats + (size_t)3 * N * HH;

    // 1. GEMM h = x @ W  (WMMA, NT column tiles/wave, pipelined K loop)
    {
        long long waves = (long long)(N / 16) * (M / (16 * NT));
        gat_gemm_wmma_f32<NT><<<cdiv_i(waves * 32, 256), 256, 0, stream>>>(
            xin, W, hbuf, N, K, M);
    }
    // 2. zero aggregation buffer
    hipMemsetAsync(aggbuf, 0, (size_t)N * M * sizeof(float), stream);
    // 3. attention coefficients + stat init
    gat_alpha_init<HH, C><<<cdiv_i((long long)N * HH, 256), 256, 0, stream>>>(
        hbuf, a_s, a_d, as, ad, mmax, denom, N);
    // 4-6. edge passes
    long long tot = (long long)(E + N) * HH;
    gat_edge_max<HH><<<cdiv_i(tot, 256), 256, 0, stream>>>(src, dst, as, ad, mmax, E, N);
    gat_edge_sum<HH><<<cdiv_i(tot, 256), 256, 0, stream>>>(src, dst, as, ad, mmax, denom, E, N);
    gat_edge_agg<HH, C><<<cdiv_i(tot, 256), 256, 0, stream>>>(src, dst, as, ad, mmax, denom,
                                                              hbuf, aggbuf, E, N);
    // 7. bias + activation (float4 lanes)
    long long total4 = (long long)N * M / 4;
    gat_bias_act<M / 4, DO_ELU><<<cdiv_i(total4, 256), 256, 0, stream>>>(
        aggbuf, bias, out, total4);
}

extern "C" void kernel_launch(void* const* d_in, const int* in_sizes, int n_in,
                              void* d_out, int out_size, void* d_ws, size_t ws_size,
                              hipStream_t stream) {
    const float* x      = (const float*)d_in[0];
    const int*   eidx   = (const int*)d_in[1];
    const float* W1     = (const float*)d_in[2];
    const float* a_src1 = (const float*)d_in[3];
    const float* a_dst1 = (const float*)d_in[4];
    const float* b1     = (const float*)d_in[5];
    const float* W2     = (const float*)d_in[6];
    const float* a_src2 = (const float*)d_in[7];
    const float* a_dst2 = (const float*)d_in[8];
    const float* b2     = (const float*)d_in[9];
    const float* W3     = (const float*)d_in[10];
    const float* a_src3 = (const float*)d_in[11];
    const float* a_dst3 = (const float*)d_in[12];
    const float* b3     = (const float*)d_in[13];

    const int N = in_sizes[0] / GAT_INCH;      // 50000
    const int E = in_sizes[1] / 2;             // 800000
    const int* src = eidx;                     // edge_index[0]
    const int* dst = eidx + E;                 // edge_index[1]

    float* ws = (float*)d_ws;
    const size_t M1 = (size_t)GAT_HEADS * GAT_HID;   // 256

    // Workspace layout (floats):
    //   A (N*256): h1, then overwritten in place by elu(agg1+b1) = x2
    //   B (N*256): agg1; after layer 1 reused for layer-2/3 tensors
    //   S (4*N*8): softmax stats, reused per layer
    float* A = ws;                              // N*256
    float* B = ws + (size_t)N * M1;             // N*256
    float* S = ws + (size_t)2 * N * M1;         // 4*N*HEADS

    // ---- Layer 1: 128 -> 8x32, ELU --------------------------------------
    run_layer<GAT_HEADS, GAT_HID, 4, 1>(stream, x, W1, a_src1, a_dst1, b1,
                                        /*h=*/A, /*agg=*/B, /*stats=*/S, /*out=*/A,
                                        N, E, src, dst, GAT_INCH);

    // ---- Layer 2: 256 -> 1x32, ELU (sub-allocate inside B) --------------
    float* h2   = B;                            // N*32
    float* agg2 = B + (size_t)N * GAT_HID;      // N*32
    float* x3   = B + (size_t)2 * N * GAT_HID;  // N*32
    run_layer<1, GAT_HID, 2, 1>(stream, A, W2, a_src2, a_dst2, b2,
                                h2, agg2, S, x3,
                                N, E, src, dst, (int)M1);

    // ---- Layer 3: 32 -> 1x64, no activation -----------------------------
    float* h3   = B + (size_t)3 * N * GAT_HID;  // N*64
    float* agg3 = h3 + (size_t)N * GAT_EMB;     // N*64
    run_layer<1, GAT_EMB, 4, 0>(stream, x3, W3, a_src3, a_dst3, b3,
                                h3, agg3, S, (float*)d_out,
                                N, E, src, dst, GAT_HID);
}